// ParityPairingPolicy_3178275799162
// MI455X (gfx1250) — compile-verified
//
#include <hip/hip_runtime.h>
#include <hip/hip_bf16.h>
#include <math.h>

#define HID 128
#define HEADS 4
#define NLAYERS 3
#define TI 8
#define TJ 8

typedef float v2f __attribute__((ext_vector_type(2)));
typedef float v8f __attribute__((ext_vector_type(8)));
typedef unsigned v4u __attribute__((ext_vector_type(4)));
typedef int v4i __attribute__((ext_vector_type(4)));
typedef int v8i __attribute__((ext_vector_type(8)));

#define HAS_TDM (__has_builtin(__builtin_amdgcn_tensor_load_to_lds) && \
                 __has_builtin(__builtin_amdgcn_s_wait_tensorcnt))

__device__ __forceinline__ v8f wmma_f32(v2f a, v2f b, v8f c) {
  // V_WMMA_F32_16X16X4_F32 : D = A(16x4,f32) x B(4x16,f32) + C(16x16,f32)
  return __builtin_amdgcn_wmma_f32_16x16x4_f32(false, a, false, b, (short)0, c,
                                               false, false);
}

#if HAS_TDM
__device__ __forceinline__ unsigned lds_addr_of(const void* p) {
  // generic -> LDS addrspace; AS(3) pointer value is the LDS byte offset
  return (unsigned)(size_t)(__attribute__((address_space(3))) const void*)p;
}
#endif

// ---------- ordered-uint float max (memset-0 == below -inf) ----------
__device__ __forceinline__ unsigned f2key(float f) {
  unsigned u = __float_as_uint(f);
  return (u & 0x80000000u) ? ~u : (u | 0x80000000u);
}
__device__ __forceinline__ float key2f(unsigned k) {
  return (k & 0x80000000u) ? __uint_as_float(k & 0x7FFFFFFFu)
                           : __uint_as_float(~k);
}

// ---------- node encoder: h = relu(x @ node_W + node_b) ----------
__global__ void k_node_encode(const float* __restrict__ x,
                              const float* __restrict__ W,
                              const float* __restrict__ b,
                              float* __restrict__ h, int N, int FIN) {
  int n = blockIdx.x;
  int d = threadIdx.x;
  float acc = b[d];
  for (int f = 0; f < FIN; ++f) acc += x[n * FIN + f] * W[f * HID + d];
  h[n * HID + d] = fmaxf(acc, 0.f);
}

// ---------- z = h[M,128] @ W[128,512]  (TDM A-tile stage + WMMA f32) ----------
// One block per 16-row M-tile; 8 waves each cover 4 of the 32 N-tiles, all
// sharing one LDS-resident A-tile loaded by the Tensor Data Mover.
__global__ void k_gemm_z(const float* __restrict__ h,
                         const float* __restrict__ W, float* __restrict__ z,
                         int M) {
  __shared__ float ta[16 * HID];  // 8 KB A-tile
  int row0 = blockIdx.x * 16;
#if HAS_TDM
  if (threadIdx.x < 32) {  // one wave issues the tensor DMA descriptor
    unsigned lds = lds_addr_of(ta);
    unsigned long long ga = (unsigned long long)(size_t)(h + (size_t)row0 * HID);
    v4u g0;
    g0.x = 1u;                                   // count=1, user descriptor
    g0.y = lds;                                  // lds_addr [63:32]
    g0.z = (unsigned)ga;                         // global_addr lo
    g0.w = (unsigned)((ga >> 32) & 0x01FFFFFFu)  // global_addr [56:32]
           | 0x80000000u;                        // type=2 ("image") [127:126]
    v8i g1;
    g1[0] = (int)(2u << 16);    // workgroup_mask=0, data_size=2 (4B)
    g1[1] = (int)(128u << 16);  // tensor_dim0 = 128 (bits 63:48)
    g1[2] = (int)(16u << 16);   // tensor_dim1 = 16  (bits 95:80)
    g1[3] = (int)(128u << 16);  // tile_dim0   = 128 (bits 127:112)
    g1[4] = 16;                 // tile_dim1   = 16  (bits 143:128)
    g1[5] = 128;                // tensor_dim0_stride = 128 (bits 191:160)
    g1[6] = 0;
    g1[7] = 0;
    v4i zz = {0, 0, 0, 0};      // groups 2/3 unused (2-D tile)
#if __clang_major__ >= 23
    v8i zz8 = {0, 0, 0, 0, 0, 0, 0, 0};
    __builtin_amdgcn_tensor_load_to_lds(g0, g1, zz, zz, zz8, 0);
#else
    __builtin_amdgcn_tensor_load_to_lds(g0, g1, zz, zz, 0);
#endif
    __builtin_amdgcn_s_wait_tensorcnt(0);
  }
  __syncthreads();
#else
  for (int idx = threadIdx.x; idx < 16 * HID / 4; idx += blockDim.x)
    ((float4*)ta)[idx] = ((const float4*)(h + (size_t)row0 * HID))[idx];
  __syncthreads();
#endif
  int wave = threadIdx.x >> 5;
  int lane = threadIdx.x & 31;
  int half = lane >> 4;
  int mn = lane & 15;
  const float* arow = ta + mn * HID;
  for (int t = 0; t < 4; ++t) {
    int nt = wave * 4 + t;  // 0..31
    int col0 = nt * 16;
    v8f c = {};
    for (int k0 = 0; k0 < HID; k0 += 4) {
      int ka = k0 + 2 * half;  // even
      __builtin_prefetch(&W[(size_t)(k0 + 4) * 512 + col0 + mn], 0, 1);
      v2f a = *(const v2f*)(arow + ka);  // LDS b64 load
      v2f b;
      b.x = W[(size_t)ka * 512 + col0 + mn];
      b.y = W[(size_t)(ka + 1) * 512 + col0 + mn];
      c = wmma_f32(a, b, c);
    }
#pragma unroll
    for (int v = 0; v < 8; ++v)
      z[(size_t)(row0 + half * 8 + v) * 512 + col0 + mn] = c[v];
  }
}

// ---------- per-(node,head) attention logits ----------
__global__ void k_att(const float* __restrict__ z,
                      const float* __restrict__ asrc,
                      const float* __restrict__ adst, float* __restrict__ es,
                      float* __restrict__ ed, int N) {
  int wid = (blockIdx.x * blockDim.x + threadIdx.x) >> 5;
  int lane = threadIdx.x & 31;
  if (wid >= N * HEADS) return;
  int n = wid >> 2, hh = wid & 3;
  const float* zp = z + (size_t)(n * HEADS + hh) * HID;
  const float* ap = asrc + hh * HID;
  const float* bp = adst + hh * HID;
  float s1 = 0.f, s2 = 0.f;
#pragma unroll
  for (int q = 0; q < 4; ++q) {
    float zv = zp[lane + 32 * q];
    s1 += zv * ap[lane + 32 * q];
    s2 += zv * bp[lane + 32 * q];
  }
  for (int m = 16; m >= 1; m >>= 1) {
    s1 += __shfl_xor(s1, m, 32);
    s2 += __shfl_xor(s2, m, 32);
  }
  if (lane == 0) {
    es[wid] = s1;
    ed[wid] = s2;
  }
}

// ---------- edge pass 1: segment max over dst ----------
__global__ void k_edge_max(const int* __restrict__ ei, int E, int N,
                           const float* __restrict__ es,
                           const float* __restrict__ ed,
                           unsigned* __restrict__ mkey) {
  int e = blockIdx.x * blockDim.x + threadIdx.x;
  int ET = E + N;
  if (e >= ET) return;
  int src = (e < E) ? ei[e] : (e - E);
  int dst = (e < E) ? ei[E + e] : (e - E);
#pragma unroll
  for (int hh = 0; hh < HEADS; ++hh) {
    float v = es[src * HEADS + hh] + ed[dst * HEADS + hh];
    v = (v > 0.f) ? v : 0.2f * v;  // leaky relu
    atomicMax(&mkey[dst * HEADS + hh], f2key(v));
  }
}

// ---------- edge pass 2: unnormalized softmax aggregate ----------
__global__ void k_edge_agg(const int* __restrict__ ei, int E, int N,
                           const float* __restrict__ es,
                           const float* __restrict__ ed,
                           const unsigned* __restrict__ mkey,
                           const float* __restrict__ z,
                           float* __restrict__ ssum, float* __restrict__ acc) {
  int e = blockIdx.x;
  int src = (e < E) ? ei[e] : (e - E);
  int dst = (e < E) ? ei[E + e] : (e - E);
  __shared__ float ps[HEADS];
  int d = threadIdx.x;
  if (d < HEADS) {
    float v = es[src * HEADS + d] + ed[dst * HEADS + d];
    v = (v > 0.f) ? v : 0.2f * v;
    float p = expf(v - key2f(mkey[dst * HEADS + d]));
    ps[d] = p;
    atomicAdd(&ssum[dst * HEADS + d], p);
  }
  __syncthreads();
#pragma unroll
  for (int hh = 0; hh < HEADS; ++hh) {
    float p = ps[hh];
    atomicAdd(&acc[(size_t)(dst * HEADS + hh) * HID + d],
              p * z[(size_t)(src * HEADS + hh) * HID + d]);
  }
}

// ---------- per-node: normalize, head-mean, +bias, LN, relu+residual ----------
__global__ void k_finalize(const float* __restrict__ acc,
                           const float* __restrict__ ssum,
                           const float* __restrict__ gb,
                           const float* __restrict__ lg,
                           const float* __restrict__ lb,
                           const float* __restrict__ hold,
                           float* __restrict__ hnew, int N) {
  int wid = (blockIdx.x * blockDim.x + threadIdx.x) >> 5;
  int lane = threadIdx.x & 31;
  if (wid >= N) return;
  int n = wid;
  float rs[HEADS];
#pragma unroll
  for (int hh = 0; hh < HEADS; ++hh) rs[hh] = 1.f / ssum[n * HEADS + hh];
  float v[4];
#pragma unroll
  for (int q = 0; q < 4; ++q) {
    int d = lane + 32 * q;
    float a = 0.f;
#pragma unroll
    for (int hh = 0; hh < HEADS; ++hh)
      a += acc[(size_t)(n * HEADS + hh) * HID + d] * rs[hh];
    v[q] = 0.25f * a + gb[d];
  }
  float sum = v[0] + v[1] + v[2] + v[3];
  for (int m = 16; m >= 1; m >>= 1) sum += __shfl_xor(sum, m, 32);
  float mu = sum * (1.f / HID);
  float var = 0.f;
#pragma unroll
  for (int q = 0; q < 4; ++q) {
    float t = v[q] - mu;
    var += t * t;
  }
  for (int m = 16; m >= 1; m >>= 1) var += __shfl_xor(var, m, 32);
  var *= (1.f / HID);
  float inv = 1.f / sqrtf(var + 1e-5f);
#pragma unroll
  for (int q = 0; q < 4; ++q) {
    int d = lane + 32 * q;
    float hn = (v[q] - mu) * inv * lg[d] + lb[d];
    hnew[(size_t)n * HID + d] = fmaxf(hn, 0.f) + hold[(size_t)n * HID + d];
  }
}

// ---------- scorer layer-1 factored: A = emb@sW1_top, B = emb@sW1_bot + sb1 ----------
__global__ void k_scorer_ab(const float* __restrict__ h,
                            const int* __restrict__ odd,
                            const float* __restrict__ sW1,
                            const float* __restrict__ sb1,
                            float* __restrict__ A, float* __restrict__ B) {
  int wave = threadIdx.x >> 5;
  int lane = threadIdx.x & 31;
  int half = lane >> 4;
  int mn = lane & 15;
  int tile = blockIdx.x * (blockDim.x >> 5) + wave;  // 32 mtiles * 16 ntiles
  int mt = tile >> 4, nt = tile & 15;
  int row0 = mt * 16, col0 = nt * 16;
  int r = odd[row0 + mn];
  const float* hrow = h + (size_t)r * HID;
  int cc = col0 + mn;  // 0..255  (tile-uniform side of the 128 split)
  int woff = (cc >= HID) ? HID : 0;
  int wcol = cc & (HID - 1);
  v8f c = {};
  for (int k0 = 0; k0 < HID; k0 += 4) {
    int ka = k0 + 2 * half;
    v2f a = *(const v2f*)(hrow + ka);
    v2f b;
    b.x = sW1[(size_t)(ka + woff) * HID + wcol];
    b.y = sW1[(size_t)(ka + 1 + woff) * HID + wcol];
    c = wmma_f32(a, b, c);
  }
#pragma unroll
  for (int v = 0; v < 8; ++v) {
    int row = row0 + half * 8 + v;
    if (cc < HID)
      A[(size_t)row * HID + cc] = c[v];
    else
      B[(size_t)row * HID + (cc - HID)] = c[v] + sb1[cc - HID];
  }
}

// ---------- scorer main: per 8x8 pair tile, fused relu->GEMM->relu->dot ----------
__global__ void k_scorer_main(const float* __restrict__ A,
                              const float* __restrict__ B,
                              const float* __restrict__ sW2,
                              const float* __restrict__ sb2,
                              const float* __restrict__ sW3,
                              const float* __restrict__ sb3,
                              float* __restrict__ sfull, int NODD) {
  __shared__ float t1[TI * TJ][HID + 1];  // pad: kill 16-way bank conflicts
  __shared__ float sacc[TI * TJ];
  int i0 = blockIdx.y * TI, j0 = blockIdx.x * TJ;
  for (int idx = threadIdx.x; idx < TI * TJ * HID; idx += blockDim.x) {
    int p = idx >> 7, d = idx & 127;
    int i = i0 + p / TJ, j = j0 + p % TJ;
    t1[p][d] = fmaxf(A[(size_t)i * HID + d] + B[(size_t)j * HID + d], 0.f);
  }
  if (threadIdx.x < TI * TJ) sacc[threadIdx.x] = sb3[0];
  __syncthreads();
  int wave = threadIdx.x >> 5;  // 8 waves, 4 tiles each (4 mtiles x 8 ntiles)
  int lane = threadIdx.x & 31;
  int half = lane >> 4, mn = lane & 15;
  for (int t = 0; t < 4; ++t) {
    int tile = wave * 4 + t;
    int mt = tile >> 3, nt = tile & 7;
    int row0 = mt * 16, col0 = nt * 16;
    v8f c = {};
    for (int k0 = 0; k0 < HID; k0 += 4) {
      int ka = k0 + 2 * half;
      v2f a;
      a.x = t1[row0 + mn][ka];
      a.y = t1[row0 + mn][ka + 1];
      v2f b;
      b.x = sW2[(size_t)ka * HID + col0 + mn];
      b.y = sW2[(size_t)(ka + 1) * HID + col0 + mn];
      c = wmma_f32(a, b, c);
    }
    float w3 = sW3[col0 + mn];
    float bb = sb2[col0 + mn];
#pragma unroll
    for (int v = 0; v < 8; ++v) {
      float t2 = fmaxf(c[v] + bb, 0.f) * w3;
      for (int m = 8; m >= 1; m >>= 1) t2 += __shfl_xor(t2, m, 32);
      if (mn == 0) atomicAdd(&sacc[row0 + half * 8 + v], t2);
    }
  }
  __syncthreads();
  if (threadIdx.x < TI * TJ) {
    int p = threadIdx.x;
    int i = i0 + p / TJ, j = j0 + p % TJ;
    sfull[(size_t)i * NODD + j] = sacc[p];
  }
}

// ---------- epilogue: U = triu(s,1); out = U + U^T ----------
__global__ void k_sym(const float* __restrict__ sfull, float* __restrict__ out,
                      int NODD) {
  int idx = blockIdx.x * blockDim.x + threadIdx.x;
  if (idx >= NODD * NODD) return;
  int i = idx / NODD, j = idx % NODD;
  out[idx] = (i < j) ? sfull[idx] : ((i > j) ? sfull[(size_t)j * NODD + i] : 0.f);
}

extern "C" void kernel_launch(void* const* d_in, const int* in_sizes, int n_in,
                              void* d_out, int out_size, void* d_ws,
                              size_t ws_size, hipStream_t stream) {
  (void)in_sizes; (void)n_in; (void)out_size; (void)ws_size;
  const float* x     = (const float*)d_in[0];
  const int*   ei    = (const int*)d_in[1];
  const int*   odd   = (const int*)d_in[2];
  const float* nodeW = (const float*)d_in[3];
  const float* nodeB = (const float*)d_in[4];
  const float* gatW  = (const float*)d_in[5];
  const float* attS  = (const float*)d_in[6];
  const float* attD  = (const float*)d_in[7];
  const float* gatB  = (const float*)d_in[8];
  const float* lnG   = (const float*)d_in[9];
  const float* lnB   = (const float*)d_in[10];
  const float* sW1   = (const float*)d_in[11];
  const float* sb1   = (const float*)d_in[12];
  const float* sW2   = (const float*)d_in[13];
  const float* sb2   = (const float*)d_in[14];
  const float* sW3   = (const float*)d_in[15];
  const float* sb3   = (const float*)d_in[16];

  const int N = 20000, E = 320000, FIN = 5, NODD = 512;

  float* ws = (float*)d_ws;
  size_t off = 0;
  float* h0 = ws + off;  off += (size_t)N * HID;
  float* h1 = ws + off;  off += (size_t)N * HID;
  float* z  = ws + off;  off += (size_t)N * HEADS * HID;
  float* acc = ws + off; off += (size_t)N * HEADS * HID;
  float* es = ws + off;  off += (size_t)N * HEADS;
  float* ed = ws + off;  off += (size_t)N * HEADS;
  unsigned* mkey = (unsigned*)(ws + off); off += (size_t)N * HEADS;
  float* ssum = ws + off; off += (size_t)N * HEADS;
  float* Abuf = ws + off; off += (size_t)NODD * HID;
  float* Bbuf = ws + off; off += (size_t)NODD * HID;
  float* sfull = ws + off; off += (size_t)NODD * NODD;

  k_node_encode<<<N, HID, 0, stream>>>(x, nodeW, nodeB, h0, N, FIN);

  float* hc = h0;
  float* hn = h1;
  for (int l = 0; l < NLAYERS; ++l) {
    hipMemsetAsync(acc, 0, (size_t)N * HEADS * HID * sizeof(float), stream);
    hipMemsetAsync(ssum, 0, (size_t)N * HEADS * sizeof(float), stream);
    hipMemsetAsync(mkey, 0, (size_t)N * HEADS * sizeof(unsigned), stream);

    k_gemm_z<<<N / 16, 256, 0, stream>>>(
        hc, gatW + (size_t)l * HID * HEADS * HID, z, N);
    k_att<<<(N * HEADS) / 4, 128, 0, stream>>>(
        z, attS + (size_t)l * HEADS * HID, attD + (size_t)l * HEADS * HID, es,
        ed, N);
    k_edge_max<<<(E + N + 255) / 256, 256, 0, stream>>>(ei, E, N, es, ed, mkey);
    k_edge_agg<<<E + N, HID, 0, stream>>>(ei, E, N, es, ed, mkey, z, ssum, acc);
    k_finalize<<<(N + 3) / 4, 128, 0, stream>>>(
        acc, ssum, gatB + (size_t)l * HID, lnG + (size_t)l * HID,
        lnB + (size_t)l * HID, hc, hn, N);
    float* t = hc; hc = hn; hn = t;
  }

  k_scorer_ab<<<64, 256, 0, stream>>>(hc, odd, sW1, sb1, Abuf, Bbuf);
  dim3 g8(NODD / TJ, NODD / TI);
  k_scorer_main<<<g8, 256, 0, stream>>>(Abuf, Bbuf, sW2, sb2, sW3, sb3, sfull,
                                        NODD);
  k_sym<<<(NODD * NODD + 255) / 256, 256, 0, stream>>>(sfull, (float*)d_out,
                                                       NODD);
}